// RotatePatchLightEncoderForSniper_52836687675687
// MI455X (gfx1250) — compile-verified
//
#include <hip/hip_runtime.h>

// ---------------------------------------------------------------------------
// Output layout (flat float32, reference return order):
//   [0,                N*4)  voxel_features   [N, 4]
//   [N*4,              N*8)  voxel_coords     [N, 4]  (ints as float values)
//   [N*8,        N*8 + V*3)  patch_center_xyz [V, 3]
//   [N*8+V*3, N*8+V*3 + B )  patch_num_list   [B]
// ---------------------------------------------------------------------------

typedef __attribute__((ext_vector_type(2))) float v2f;
typedef __attribute__((ext_vector_type(8))) float v8f;

#define MAX_PTS 32
#define NCH 4

// One wave32 computes the 32-point mean for a tile of 16 voxels using
// chained V_WMMA_F32_16X16X4_F32:  D += A(16x4) x B(4x16)
//   A chunk pt = rows -> voxels, K=0..3 -> channels of point `pt`
//   B          = 4x16 identity-padded selection matrix (constant)
// so column c of D accumulates sum over points of channel c.
__global__ void __launch_bounds__(256)
voxel_mean_wmma_kernel(const float* __restrict__ pv,
                       const int* __restrict__ npts,
                       float* __restrict__ out_feat,
                       int n_voxels) {
  const int lane  = threadIdx.x & 31;
  const int wave  = (int)((blockIdx.x * blockDim.x + threadIdx.x) >> 5);
  const int vbase = wave * 16;
  if (vbase >= n_voxels) return;            // wave-uniform: EXEC stays all-ones

  const int row  = lane & 15;               // voxel within tile (A-matrix M)
  const int half = lane >> 4;               // 0 -> supplies K=0,1 ; 1 -> K=2,3

  // A-matrix 16x4 f32 layout: VGPR0 = {K=0 | K=2}, VGPR1 = {K=1 | K=3}.
  // Lane L loads float2 {ch(2*half), ch(2*half+1)} of each point:
  //   .x lands in A-VGPR0, .y in A-VGPR1.
  const v2f* __restrict__ src =
      (const v2f*)pv + (size_t)(vbase + row) * (MAX_PTS * NCH / 2) + half;

  // B-matrix 4x16 f32 (mirrors A layout): VGPR0 rows K=0 (lanes 0-15) and
  // K=2 (lanes 16-31); VGPR1 rows K=1 and K=3. Identity => B[k][n] = (n==k).
  v2f b;
  b.x = (lane == 0 || lane == 18) ? 1.0f : 0.0f;
  b.y = (lane == 1 || lane == 19) ? 1.0f : 0.0f;

  v8f acc = {};
#pragma unroll
  for (int pt = 0; pt < MAX_PTS; ++pt) {
    v2f a = src[pt * 2];                    // global_load_b64, imm offset pt*16
    acc = __builtin_amdgcn_wmma_f32_16x16x4_f32(
        /*neg_a=*/false, a, /*neg_b=*/false, b,
        /*c_mod=*/(short)0, acc, /*reuse_a=*/false, /*reuse_b=*/false);
  }

  // D 16x16 f32 layout: VGPR i, lanes 0-15 -> D[i][n]; lanes 16-31 -> D[i+8][n].
  // Only columns n=0..3 are meaningful -> 8 lanes write everything.
  const int c = lane & 15;
  if (c < NCH) {
    const int rbase = (lane >> 4) * 8;
#pragma unroll
    for (int i = 0; i < 8; ++i) {
      const int v = vbase + rbase + i;
      const int n = npts[v];
      const float norm = (float)(n < 1 ? 1 : n);
      out_feat[v * NCH + c] = acc[i] / norm;
    }
  }
}

// Single-block inclusive scan of the per-patch voxel counts (P <= 2048).
__global__ void __launch_bounds__(256)
scan_counts_kernel(const int* __restrict__ counts, int* __restrict__ cum, int P) {
  __shared__ int tsum[256];
  const int t   = threadIdx.x;
  const int per = (P + 255) / 256;

  int s = 0;
  for (int i = 0; i < per; ++i) {
    const int idx = t * per + i;
    if (idx < P) s += counts[idx];
  }
  tsum[t] = s;
  __syncthreads();
  for (int off = 1; off < 256; off <<= 1) {     // Hillis-Steele inclusive scan
    const int v = (t >= off) ? tsum[t - off] : 0;
    __syncthreads();
    tsum[t] += v;
    __syncthreads();
  }
  int run = tsum[t] - s;                        // exclusive prefix of this thread
  for (int i = 0; i < per; ++i) {
    const int idx = t * per + i;
    if (idx < P) { run += counts[idx]; cum[idx] = run; }
  }
}

// voxel_coords: binary search the (LDS-staged) cumsum per voxel, replace col 0.
__global__ void __launch_bounds__(256)
coords_kernel(const int* __restrict__ coords, const int* __restrict__ cum,
              float* __restrict__ out, int n_voxels, int P) {
  __shared__ int scum[2048];
  for (int i = threadIdx.x; i < P; i += blockDim.x) scum[i] = cum[i];
  __syncthreads();

  const int j = blockIdx.x * blockDim.x + threadIdx.x;
  if (j >= n_voxels) return;

  // searchsorted(..., side='right'): first i with cum[i] > j
  int lo = 0, hi = P;
  while (lo < hi) {
    const int mid = (lo + hi) >> 1;
    if (scum[mid] <= j) lo = mid + 1; else hi = mid;
  }
  out[j * 4 + 0] = (float)lo;
  out[j * 4 + 1] = (float)coords[j * 4 + 1];
  out[j * 4 + 2] = (float)coords[j * 4 + 2];
  out[j * 4 + 3] = (float)coords[j * 4 + 3];
}

// patch_center_xyz (z-padded) + patch_num_list.
__global__ void __launch_bounds__(256)
tail_kernel(const float* __restrict__ centers, const int* __restrict__ nums,
            float* __restrict__ out_c, float* __restrict__ out_n, int V, int B) {
  const int i = blockIdx.x * blockDim.x + threadIdx.x;
  if (i < V) {
    out_c[i * 3 + 0] = centers[i * 2 + 0];
    out_c[i * 3 + 1] = centers[i * 2 + 1];
    out_c[i * 3 + 2] = 0.0f;
  }
  if (i < B) out_n[i] = (float)nums[i];
}

extern "C" void kernel_launch(void* const* d_in, const int* in_sizes, int n_in,
                              void* d_out, int out_size, void* d_ws, size_t ws_size,
                              hipStream_t stream) {
  const float* pv      = (const float*)d_in[0];   // [N, 32, 4] f32
  const int*   npts    = (const int*)  d_in[1];   // [N] i32
  const int*   coords  = (const int*)  d_in[2];   // [N, 4] i32
  const int*   pcounts = (const int*)  d_in[3];   // [P] i32
  const float* centers = (const float*)d_in[4];   // [V, 2] f32
  const int*   vnums   = (const int*)  d_in[5];   // [B] i32

  const int N = in_sizes[1];
  const int P = in_sizes[3];
  const int V = in_sizes[4] / 2;
  const int B = in_sizes[5];

  float* out_feat    = (float*)d_out;
  float* out_coords  = out_feat    + (size_t)N * 4;
  float* out_centers = out_coords  + (size_t)N * 4;
  float* out_nums    = out_centers + (size_t)V * 3;

  int* cum = (int*)d_ws;                          // P ints of scratch

  voxel_mean_wmma_kernel<<<(N + 127) / 128, 256, 0, stream>>>(pv, npts, out_feat, N);
  scan_counts_kernel<<<1, 256, 0, stream>>>(pcounts, cum, P);
  coords_kernel<<<(N + 255) / 256, 256, 0, stream>>>(coords, cum, out_coords, N, P);
  tail_kernel<<<(V + 255) / 256, 256, 0, stream>>>(centers, vnums, out_centers, out_nums, V, B);
}